// MultiKernelMGPLayer_18116172055215
// MI455X (gfx1250) — compile-verified
//
#include <hip/hip_runtime.h>
#include <math.h>

// Problem constants (match reference)
#define BB      64
#define NMAXC   512
#define GC      64
#define FC      16
#define SC      10
#define MC      (GC*FC)     // 1024
#define ADD_DIAG_F 1e-3f
#define KB      32          // blocking factor for chol/trsm

typedef __attribute__((ext_vector_type(2))) float v2f;
typedef __attribute__((ext_vector_type(8))) float v8f;

// ---- workspace layout (float offsets) ----
#define WS_KDV    0          // F*F
#define WS_KDL    256        // F*F
#define WS_NOISE  512        // F
#define WS_INVLV  528
#define WS_INVLL  529
#define WS_SAMPLE_BASE 1024ull
#define OFF_TBIG  0ull
#define OFF_YR    512ull
#define OFF_ALPHA 1024ull
#define OFF_SIGMA 1536ull                         // NMAX*NMAX = 262144
#define OFF_KXT   (OFF_SIGMA + 262144ull)         // M*NMAX    = 524288
#define OFF_W     (OFF_KXT   + 524288ull)         // NMAX*M    = 524288  (W = L^-1 K_XT^T)
#define OFF_SIG   (OFF_W     + 524288ull)         // M*M       = 1048576
#define PS        (OFF_SIG   + 1048576ull)        // floats/sample

#define NTILE_LT  528                             // lower-tri 32x32 tiles of M=1024

#define WMMA_F32(A, B, C) \
  __builtin_amdgcn_wmma_f32_16x16x4_f32(false, (A), false, (B), (short)0, (C), false, false)

// map lower-triangle linear tile index -> (tr,tc), tr >= tc (wave-uniform)
__device__ __forceinline__ void lt_map(int t, int& tr, int& tc) {
  tr = (int)((sqrtf(8.0f*(float)t + 1.0f) - 1.0f) * 0.5f);
  while (((tr + 1) * (tr + 2)) >> 1 <= t) ++tr;
  while ((tr * (tr + 1)) >> 1 > t) --tr;
  tc = t - ((tr * (tr + 1)) >> 1);
}

// ------------------------------------------------------------------
__global__ void prep_globals(const float* __restrict__ kdhv, const float* __restrict__ kdhl,
                             const float* __restrict__ log_noises,
                             const float* __restrict__ llv, const float* __restrict__ lll,
                             float* __restrict__ ws) {
  int tid = threadIdx.x;            // 256 threads == F*F
  int i = tid / FC, j = tid % FC;
  float sv = 0.f, sl = 0.f;
  int kmax = (i < j) ? i : j;
  for (int k = 0; k <= kmax; ++k) {
    sv += kdhv[i*FC + k] * kdhv[j*FC + k];
    sl += kdhl[i*FC + k] * kdhl[j*FC + k];
  }
  ws[WS_KDV + i*FC + j] = sv;
  ws[WS_KDL + i*FC + j] = sl;
  if (tid < FC) ws[WS_NOISE + tid] = __expf(log_noises[tid]);
  if (tid == 0) { ws[WS_INVLV] = __expf(-llv[0]); ws[WS_INVLL] = __expf(-lll[0]); }
}

// ------------------------------------------------------------------
__global__ void prep_sample(const float* __restrict__ Y, const float* __restrict__ T,
                            const int* __restrict__ indT, const int* __restrict__ num_obs,
                            float* __restrict__ ws) {
  int b = blockIdx.x;
  float* wsb = ws + WS_SAMPLE_BASE + (size_t)b * PS;
  int nobs = num_obs[b];
  for (int n = threadIdx.x; n < NMAXC; n += blockDim.x) {
    int it = indT[b*NMAXC + n];
    wsb[OFF_TBIG + n] = T[b*NMAXC + it];
    wsb[OFF_YR   + n] = (n < nobs) ? Y[b*NMAXC + it] : 0.f;
  }
}

// ------------------------------------------------------------------
__global__ void build_sigma(const int* __restrict__ indKD, const int* __restrict__ num_obs,
                            float* __restrict__ ws) {
  int b = blockIdx.x;
  float* wsb = ws + WS_SAMPLE_BASE + (size_t)b * PS;
  int nobs = num_obs[b];
  float inv_lv = ws[WS_INVLV], inv_ll = ws[WS_INVLL];
  int e = blockIdx.y * blockDim.x + threadIdx.x;   // < 262144
  int i = e >> 9, j = e & 511;
  int di = indKD[b*NMAXC + i], dj = indKD[b*NMAXC + j];
  float ad = fabsf(wsb[OFF_TBIG + i] - wsb[OFF_TBIG + j]);
  float kp = ws[WS_KDV + di*FC + dj] * __expf(-ad*inv_lv)
           + ws[WS_KDL + di*FC + dj] * __expf(-ad*inv_ll);
  if (i == j) kp += ws[WS_NOISE + di];
  bool mi = i < nobs, mj = j < nobs;
  float v = (mi && mj) ? kp : 0.f;
  if (i == j) v += mi ? ADD_DIAG_F : 1.f;
  wsb[OFF_SIGMA + (size_t)i*NMAXC + j] = v;
}

// ------------------------------------------------------------------
// Blocked right-looking Cholesky (lower), one workgroup (8 waves) per sample.
// Diagonal block in LDS, panel solve row-parallel, trailing SYRK on WMMA.
__global__ void cholesky_blocked(float* __restrict__ ws, unsigned long long off, int N) {
  __shared__ float D[KB*KB];
  int b = blockIdx.x;
  float* A = ws + WS_SAMPLE_BASE + (size_t)b * PS + off;
  int tid = threadIdx.x;
  int lane = tid & 31, wid = tid >> 5;
  int half = lane >> 4, lm = lane & 15;

  for (int k0 = 0; k0 < N; k0 += KB) {
    // load diagonal block to LDS
    for (int e = tid; e < KB*KB; e += 256) {
      int i = e >> 5, j = e & 31;
      D[e] = A[(size_t)(k0 + i)*N + (k0 + j)];
    }
    __syncthreads();
    // factor 32x32 in LDS
    for (int kk = 0; kk < KB; ++kk) {
      if (tid == 0) D[kk*KB + kk] = sqrtf(fmaxf(D[kk*KB + kk], 1e-20f));
      __syncthreads();
      float dk = D[kk*KB + kk];
      if (tid > kk && tid < KB) D[tid*KB + kk] /= dk;
      __syncthreads();
      if (tid > kk && tid < KB) {
        float lik = D[tid*KB + kk];
        for (int j = kk + 1; j <= tid; ++j)
          D[tid*KB + j] -= lik * D[j*KB + kk];
      }
      __syncthreads();
    }
    // write back diagonal block (lower)
    for (int e = tid; e < KB*KB; e += 256) {
      int i = e >> 5, j = e & 31;
      if (j <= i) A[(size_t)(k0 + i)*N + (k0 + j)] = D[e];
    }
    __syncthreads();
    // panel solve: rows below, row_i <- row_i * D^{-T}
    for (int i = k0 + KB + tid; i < N; i += 256) {
      float* Ai = A + (size_t)i*N + k0;
      for (int j = 0; j < KB; ++j) {
        float acc = Ai[j];
        for (int t = 0; t < j; ++t) acc -= Ai[t] * D[j*KB + t];
        Ai[j] = acc / D[j*KB + j];
      }
    }
    __syncthreads();
    // trailing SYRK: A22(lower) -= P P^T, P = panel rows, on WMMA
    int rem = N - k0 - KB;
    if (rem > 0) {
      int nb = rem >> 5;
      int ntile = (nb * (nb + 1)) >> 1;
      for (int t = wid; t < ntile; t += 8) {       // wave-uniform loop
        int tr, tc; lt_map(t, tr, tc);
        int r0 = k0 + KB + (tr << 5);
        int c0 = k0 + KB + (tc << 5);
        v8f a00 = {}, a01 = {}, a10 = {}, a11 = {};
        for (int kk = 0; kk < KB; kk += 4) {
          int kb = kk + half*2;
          v2f pa0 = *(const v2f*)&A[(size_t)(r0      + lm)*N + k0 + kb];
          v2f pa1 = *(const v2f*)&A[(size_t)(r0 + 16 + lm)*N + k0 + kb];
          v2f pb0 = *(const v2f*)&A[(size_t)(c0      + lm)*N + k0 + kb];
          v2f pb1 = *(const v2f*)&A[(size_t)(c0 + 16 + lm)*N + k0 + kb];
          a00 = WMMA_F32(pa0, pb0, a00);
          a01 = WMMA_F32(pa0, pb1, a01);
          a10 = WMMA_F32(pa1, pb0, a10);
          a11 = WMMA_F32(pa1, pb1, a11);
        }
        #pragma unroll
        for (int r = 0; r < 8; ++r) {
          int m0 = r0 + r + half*8, m1 = m0 + 16;
          int n0 = c0 + lm,         n1 = n0 + 16;
          A[(size_t)m0*N + n0] -= a00[r];
          A[(size_t)m0*N + n1] -= a01[r];
          A[(size_t)m1*N + n0] -= a10[r];
          A[(size_t)m1*N + n1] -= a11[r];
        }
      }
    }
    __syncthreads();
  }
}

// ------------------------------------------------------------------
// alpha = (L L^T)^{-1} Yr   via LDS-reduced fwd/bwd substitution
__global__ void solve_alpha(float* __restrict__ ws) {
  __shared__ float y[NMAXC];
  __shared__ float red[256];
  int b = blockIdx.x;
  float* wsb = ws + WS_SAMPLE_BASE + (size_t)b * PS;
  const float* L = wsb + OFF_SIGMA;
  int tid = threadIdx.x;
  for (int n = tid; n < NMAXC; n += 256) y[n] = wsb[OFF_YR + n];
  __syncthreads();
  for (int i = 0; i < NMAXC; ++i) {
    float p = 0.f;
    for (int j = tid; j < i; j += 256) p += L[(size_t)i*NMAXC + j] * y[j];
    red[tid] = p; __syncthreads();
    for (int s = 128; s > 0; s >>= 1) { if (tid < s) red[tid] += red[tid + s]; __syncthreads(); }
    if (tid == 0) y[i] = (y[i] - red[0]) / L[(size_t)i*NMAXC + i];
    __syncthreads();
  }
  for (int i = NMAXC - 1; i >= 0; --i) {
    float p = 0.f;
    for (int j = i + 1 + tid; j < NMAXC; j += 256) p += L[(size_t)j*NMAXC + i] * y[j];
    red[tid] = p; __syncthreads();
    for (int s = 128; s > 0; s >>= 1) { if (tid < s) red[tid] += red[tid + s]; __syncthreads(); }
    if (tid == 0) y[i] = (y[i] - red[0]) / L[(size_t)i*NMAXC + i];
    __syncthreads();
  }
  for (int n = tid; n < NMAXC; n += 256) wsb[OFF_ALPHA + n] = y[n];
}

// ------------------------------------------------------------------
__global__ void build_kxt_kxi(const float* __restrict__ X, const int* __restrict__ indKD,
                              const int* __restrict__ num_obs, float* __restrict__ ws) {
  int b = blockIdx.x;
  float* wsb = ws + WS_SAMPLE_BASE + (size_t)b * PS;
  float inv_lv = ws[WS_INVLV], inv_ll = ws[WS_INVLL];
  long e = (long)blockIdx.y * blockDim.x + threadIdx.x;   // < 1572864
  if (e < (long)MC * NMAXC) {
    int m = (int)(e >> 9), n = (int)(e & 511);
    int dm = m >> 6;
    int dn = indKD[b*NMAXC + n];
    float ad = fabsf(X[b*GC + (m & 63)] - wsb[OFF_TBIG + n]);
    float v = ws[WS_KDV + dm*FC + dn] * __expf(-ad*inv_lv)
            + ws[WS_KDL + dm*FC + dn] * __expf(-ad*inv_ll);
    wsb[OFF_KXT + (size_t)m*NMAXC + n] = (n < num_obs[b]) ? v : 0.f;
  } else {
    long e2 = e - (long)MC * NMAXC;
    int m = (int)(e2 >> 10), n = (int)(e2 & 1023);
    int dm = m >> 6, dn = n >> 6;
    float ad = fabsf(X[b*GC + (m & 63)] - X[b*GC + (n & 63)]);
    float v = ws[WS_KDV + dm*FC + dn] * __expf(-ad*inv_lv)
            + ws[WS_KDL + dm*FC + dn] * __expf(-ad*inv_ll);
    wsb[OFF_SIG + (size_t)m*MC + n] = v;
  }
}

// ------------------------------------------------------------------
// Blocked forward TRSM on WMMA: W = L^{-1} K_XT^T, block-rows of 32.
// Update U = RHS - L[I,0:k0] @ W[0:k0,:] runs on WMMA; 32x32 diag solve
// per column against LDS-resident diagonal block.
__global__ void trsm_blocked(float* __restrict__ ws) {
  __shared__ float D[KB*KB];
  int b = blockIdx.x;
  float* wsb = ws + WS_SAMPLE_BASE + (size_t)b * PS;
  const float* L   = wsb + OFF_SIGMA;
  const float* KXT = wsb + OFF_KXT;
  float* W = wsb + OFF_W;
  int tid = threadIdx.x;
  int lane = tid & 31, wid = tid >> 5;
  int half = lane >> 4, lm = lane & 15;

  for (int k0 = 0; k0 < NMAXC; k0 += KB) {
    // load diagonal block of L
    for (int e = tid; e < KB*KB; e += 256) {
      int i = e >> 5, j = e & 31;
      D[e] = L[(size_t)(k0 + i)*NMAXC + (k0 + j)];
    }
    __syncthreads();
    // GEMM update: W[k0+j][c] = KXT[c][k0+j] - sum_{t<k0} L[k0+j][t] W[t][c]
    for (int ct = wid; ct < MC/KB; ct += 8) {      // wave-uniform col tiles
      int c0 = ct << 5;
      v8f a00 = {}, a01 = {}, a10 = {}, a11 = {};
      for (int kt = 0; kt < k0; kt += 4) {
        int kb = kt + half*2;
        v2f la0 = *(const v2f*)&L[(size_t)(k0      + lm)*NMAXC + kb];
        v2f la1 = *(const v2f*)&L[(size_t)(k0 + 16 + lm)*NMAXC + kb];
        v2f wb0, wb1;
        wb0.x = W[(size_t) kb     *MC + c0 + lm];
        wb0.y = W[(size_t)(kb + 1)*MC + c0 + lm];
        wb1.x = W[(size_t) kb     *MC + c0 + 16 + lm];
        wb1.y = W[(size_t)(kb + 1)*MC + c0 + 16 + lm];
        a00 = WMMA_F32(la0, wb0, a00);
        a01 = WMMA_F32(la0, wb1, a01);
        a10 = WMMA_F32(la1, wb0, a10);
        a11 = WMMA_F32(la1, wb1, a11);
      }
      #pragma unroll
      for (int r = 0; r < 8; ++r) {
        int j0 = r + half*8, j1 = j0 + 16;
        int n0 = c0 + lm,    n1 = n0 + 16;
        W[(size_t)(k0 + j0)*MC + n0] = KXT[(size_t)n0*NMAXC + k0 + j0] - a00[r];
        W[(size_t)(k0 + j0)*MC + n1] = KXT[(size_t)n1*NMAXC + k0 + j0] - a01[r];
        W[(size_t)(k0 + j1)*MC + n0] = KXT[(size_t)n0*NMAXC + k0 + j1] - a10[r];
        W[(size_t)(k0 + j1)*MC + n1] = KXT[(size_t)n1*NMAXC + k0 + j1] - a11[r];
      }
    }
    __syncthreads();
    // triangular solve of the 32-row block, 4 columns per thread
    for (int c = tid; c < MC; c += 256) {
      for (int j = 0; j < KB; ++j) {
        float acc = W[(size_t)(k0 + j)*MC + c];
        for (int t = 0; t < j; ++t) acc -= D[j*KB + t] * W[(size_t)(k0 + t)*MC + c];
        W[(size_t)(k0 + j)*MC + c] = acc / D[j*KB + j];
      }
    }
    __syncthreads();
  }
}

// ------------------------------------------------------------------
// SYRK: Sig(lower) = mask( K_Xi - W^T W ) + diag ; 32x32 tile per wave.
__device__ __forceinline__ void write_sig(float* __restrict__ Sig, int m, int n,
                                          float acc, int xlen) {
  float s = Sig[(size_t)m*MC + n] - acc;           // K_Xi - W^T W
  bool gm = (m & 63) < xlen, gn = (n & 63) < xlen;
  float v = (gm && gn) ? s : 0.f;
  if (m == n) v += gm ? ADD_DIAG_F : 1.f;
  Sig[(size_t)m*MC + n] = v;
}

__global__ void syrk_sig(const int* __restrict__ num_x, float* __restrict__ ws) {
  int lane = threadIdx.x & 31;
  int wid  = threadIdx.x >> 5;
  int gt = blockIdx.x * 8 + wid;                   // NTILE_LT*BB waves total
  int b  = gt / NTILE_LT;
  int t  = gt - b * NTILE_LT;
  int tr, tc; lt_map(t, tr, tc);
  int row0 = tr << 5, col0 = tc << 5;

  float* wsb = ws + WS_SAMPLE_BASE + (size_t)b * PS;
  const float* W = wsb + OFF_W;                    // [NMAX x M], k-major
  float* Sig = wsb + OFF_SIG;
  int xlen = num_x[b];

  int half = lane >> 4, lm = lane & 15;
  v8f acc00 = {}, acc01 = {}, acc10 = {}, acc11 = {};
  for (int kk = 0; kk < NMAXC; kk += 4) {
    int kb = kk + half*2;
    const float* Wk0 = W + (size_t)kb * MC;
    const float* Wk1 = Wk0 + MC;
    __builtin_prefetch(Wk0 + 8*MC + row0 + lm, 0, 3);
    __builtin_prefetch(Wk0 + 8*MC + col0 + lm, 0, 3);
    v2f a0, a1, b0, b1;                            // A[m][k] = W[k][m] (W^T)
    a0.x = Wk0[row0 + lm];       a0.y = Wk1[row0 + lm];
    a1.x = Wk0[row0 + 16 + lm];  a1.y = Wk1[row0 + 16 + lm];
    b0.x = Wk0[col0 + lm];       b0.y = Wk1[col0 + lm];
    b1.x = Wk0[col0 + 16 + lm];  b1.y = Wk1[col0 + 16 + lm];
    acc00 = WMMA_F32(a0, b0, acc00);
    acc01 = WMMA_F32(a0, b1, acc01);
    acc10 = WMMA_F32(a1, b0, acc10);
    acc11 = WMMA_F32(a1, b1, acc11);
  }
  #pragma unroll
  for (int r = 0; r < 8; ++r) {
    int m0 = row0 + r + half*8, m1 = m0 + 16;
    int n0 = col0 + lm,         n1 = n0 + 16;
    write_sig(Sig, m0, n0, acc00[r], xlen);
    write_sig(Sig, m0, n1, acc01[r], xlen);
    write_sig(Sig, m1, n0, acc10[r], xlen);
    write_sig(Sig, m1, n1, acc11[r], xlen);
  }
}

// ------------------------------------------------------------------
// out = roll/mask( cS @ eps + Mu )  — Mu computed inline; cS lower only.
__global__ void draws_out(const float* __restrict__ eps, const int* __restrict__ num_x,
                          const float* __restrict__ ws, float* __restrict__ out) {
  int b = blockIdx.x;
  const float* wsb   = ws + WS_SAMPLE_BASE + (size_t)b * PS;
  const float* cS    = wsb + OFF_SIG;
  const float* KXT   = wsb + OFF_KXT;
  const float* alpha = wsb + OFF_ALPHA;
  int m = blockIdx.y * blockDim.x + threadIdx.x;   // 0..1023
  int xlen = num_x[b];
  float mu = 0.f;
  for (int j = 0; j < NMAXC; ++j) mu += KXT[(size_t)m*NMAXC + j] * alpha[j];
  float acc[SC];
  #pragma unroll
  for (int s = 0; s < SC; ++s) acc[s] = mu;
  const float* epsb = eps + (size_t)b * MC * SC;
  for (int j = 0; j <= m; ++j) {
    float cv = cS[(size_t)m*MC + j];
    const float* ej = epsb + (size_t)j * SC;
    #pragma unroll
    for (int s = 0; s < SC; ++s) acc[s] += cv * ej[s];
  }
  int f = m >> 6, gsrc = m & 63;
  bool valid = gsrc < xlen;
  int gdst = gsrc + (GC - xlen); if (gdst >= GC) gdst -= GC;   // jnp.roll
  #pragma unroll
  for (int s = 0; s < SC; ++s)
    out[(((size_t)(b*SC + s))*GC + gdst)*FC + f] = valid ? acc[s] : 0.f;
}

// ------------------------------------------------------------------
extern "C" void kernel_launch(void* const* d_in, const int* in_sizes, int n_in,
                              void* d_out, int out_size, void* d_ws, size_t ws_size,
                              hipStream_t stream) {
  const float* Y          = (const float*)d_in[0];
  const float* T          = (const float*)d_in[1];
  const float* X          = (const float*)d_in[2];
  const float* eps        = (const float*)d_in[3];
  const float* KDhv       = (const float*)d_in[4];
  const float* KDhl       = (const float*)d_in[5];
  const float* log_noises = (const float*)d_in[6];
  const float* llv        = (const float*)d_in[7];
  const float* lll        = (const float*)d_in[8];
  const int*   indKD      = (const int*)d_in[9];
  const int*   indT       = (const int*)d_in[10];
  const int*   num_obs    = (const int*)d_in[11];
  const int*   num_x      = (const int*)d_in[12];
  float* ws  = (float*)d_ws;
  float* out = (float*)d_out;

  prep_globals    <<<1, 256, 0, stream>>>(KDhv, KDhl, log_noises, llv, lll, ws);
  prep_sample     <<<BB, 256, 0, stream>>>(Y, T, indT, num_obs, ws);
  build_sigma     <<<dim3(BB, 1024), 256, 0, stream>>>(indKD, num_obs, ws);
  cholesky_blocked<<<BB, 256, 0, stream>>>(ws, OFF_SIGMA, NMAXC);
  solve_alpha     <<<BB, 256, 0, stream>>>(ws);
  build_kxt_kxi   <<<dim3(BB, 6144), 256, 0, stream>>>(X, indKD, num_obs, ws);
  trsm_blocked    <<<BB, 256, 0, stream>>>(ws);
  syrk_sig        <<<(NTILE_LT * BB) / 8, 256, 0, stream>>>(num_x, ws);
  cholesky_blocked<<<BB, 256, 0, stream>>>(ws, OFF_SIG, MC);
  draws_out       <<<dim3(BB, 4), 256, 0, stream>>>(eps, num_x, ws, out);
}